// SlaterDetSampler_ordered_46308337385550
// MI455X (gfx1250) — compile-verified
//
#include <hip/hip_runtime.h>
#include <math.h>

#define D 384
#define NN 96
#define NWAVES (D / 32)

typedef __attribute__((ext_vector_type(2))) float v2f;
typedef __attribute__((ext_vector_type(8))) float v8f;

// ---------------------------------------------------------------------------
// Kernel 1: G = I - P P^T  with P = eigfunc[:, :96]  (fp32 WMMA 16x16x4)
// One wave per 16x16 output tile; K-loop of 24 WMMA ops.
// ---------------------------------------------------------------------------
__global__ __launch_bounds__(32) void build_G_kernel(const float* __restrict__ eig,
                                                     float* __restrict__ G) {
  const int i0 = blockIdx.x * 16;
  const int j0 = blockIdx.y * 16;
  const int lane = threadIdx.x & 31;
  const int mr = lane & 15;          // row (A) / col (B) within tile
  const int kb = (lane >> 4) << 1;   // 0 or 2
  const float* rowA = eig + (size_t)(i0 + mr) * D;
  const float* rowB = eig + (size_t)(j0 + mr) * D;
  v8f c = {};
  for (int k = 0; k < NN; k += 4) {
    v2f a, b;
    a.x = rowA[k + kb];
    a.y = rowA[k + kb + 1];
    b.x = rowB[k + kb];
    b.y = rowB[k + kb + 1];
    c = __builtin_amdgcn_wmma_f32_16x16x4_f32(false, a, false, b, (short)0, c,
                                              false, false);
  }
  const int half = lane >> 4;
#pragma unroll
  for (int r = 0; r < 8; ++r) {
    const int gi = i0 + r + half * 8;
    const int gj = j0 + mr;
    G[gi * D + gj] = ((gi == gj) ? 1.0f : 0.0f) - c[r];
  }
}

// ---------------------------------------------------------------------------
// Scalar single-column elimination (head/tail columns of a pass).
// ---------------------------------------------------------------------------
__device__ __forceinline__ void eliminate_one_col(float* __restrict__ A, int c,
                                                  float* __restrict__ pivs,
                                                  int tid, float* s_inv) {
  if (tid == 0) {
    const float piv = A[c * D + c];
    if (pivs) pivs[c] = piv;
    *s_inv = (fabsf(piv) > 1e-30f) ? (1.0f / piv) : 0.0f;
  }
  __syncthreads();
  const float inv = *s_inv;
  const int n = D - 1 - c;
  if (tid < n) {
    const int j = c + 1 + tid;
    const float urow = A[c * D + j];
    for (int r = c + 1; r < D; ++r) {
      const float f = A[r * D + c] * inv;
      A[r * D + j] -= f * urow;
    }
  }
  __syncthreads();
}

// ---------------------------------------------------------------------------
// Blocked step, NB=16, c must be 16-aligned.  Factors columns c..c+15 and
// applies the trailing update A22 -= L21 * U12 with K=16 via 4 chained
// v_wmma_f32_16x16x4_f32 per 16x16 tile (trailing region stays 16-aligned,
// D % 16 == 0 -> no fringe, EXEC all-ones for every WMMA).
// ---------------------------------------------------------------------------
__device__ __forceinline__ void eliminate_panel16(float* __restrict__ A, int c,
                                                  float* __restrict__ pivs,
                                                  int tid, float* s_invs,
                                                  float (*s_Lm)[16]) {
  // ---- Phase A: factor panel (updates restricted to panel columns) --------
  for (int t = 0; t < 16; ++t) {
    const int cc = c + t;
    if (tid == 0) {
      const float piv = A[cc * D + cc];
      if (pivs) pivs[cc] = piv;
      s_invs[t] = (fabsf(piv) > 1e-30f) ? (1.0f / piv) : 0.0f;
    }
    __syncthreads();
    const float inv = s_invs[t];
    const int nr = D - 1 - cc;
    if (tid < nr) {
      const int r = cc + 1 + tid;
      const float f = A[r * D + cc] * inv;
      for (int j = cc + 1; j <= c + 15; ++j) A[r * D + j] -= f * A[cc * D + j];
    }
    __syncthreads();
  }
  // ---- Multipliers for the panel-row triangular solve ---------------------
  if (tid < 256) {
    const int i = tid >> 4, t = tid & 15;
    s_Lm[i][t] = (t < i) ? A[(c + i) * D + (c + t)] * s_invs[t] : 0.0f;
  }
  __syncthreads();
  // ---- Phase B: U12 = L11^-1 * A12, one thread per trailing column --------
  const int ncols = D - c - 16;
  if (tid < ncols) {
    const int j = c + 16 + tid;
    float w[16];
#pragma unroll
    for (int t = 0; t < 16; ++t) w[t] = A[(c + t) * D + j];
#pragma unroll
    for (int i = 1; i < 16; ++i) {
      float s = w[i];
#pragma unroll
      for (int t = 0; t < i; ++t) s -= s_Lm[i][t] * w[t];
      w[i] = s;
      A[(c + i) * D + j] = s;
    }
  }
  __syncthreads();
  // ---- Phase C: trailing GEMM update via WMMA (K=16) ----------------------
  const int nt = (D - c - 16) >> 4;
  const int wv = tid >> 5;
  const int lane = tid & 31;
  const int mn = lane & 15;   // row of A-tile / col of B- and C-tiles
  const int half = lane >> 4;
  const int kb = half << 1;
  for (int tile = wv; tile < nt * nt; tile += NWAVES) {
    const int tr = c + 16 + (tile / nt) * 16;
    const int tc = c + 16 + (tile % nt) * 16;
    v8f acc;
#pragma unroll
    for (int r = 0; r < 8; ++r) acc[r] = A[(tr + r + half * 8) * D + tc + mn];
#pragma unroll
    for (int kk = 0; kk < 4; ++kk) {
      const int t0 = kk * 4 + kb;
      v2f a, b;
      // A operand: -L21 tile (sign + 1/pivot folded in)
      a.x = A[(tr + mn) * D + (c + t0)] * (-s_invs[t0]);
      a.y = A[(tr + mn) * D + (c + t0 + 1)] * (-s_invs[t0 + 1]);
      // B operand: U12 rows
      b.x = A[(c + t0) * D + tc + mn];
      b.y = A[(c + t0 + 1) * D + tc + mn];
      acc = __builtin_amdgcn_wmma_f32_16x16x4_f32(false, a, false, b, (short)0,
                                                  acc, false, false);
    }
#pragma unroll
    for (int r = 0; r < 8; ++r) A[(tr + r + half * 8) * D + tc + mn] = acc[r];
  }
  __syncthreads();
}

// ---------------------------------------------------------------------------
// Eliminate columns [cstart, cendExcl): scalar until 16-aligned, blocked WMMA
// panels while >=16 columns remain, scalar tail.
// ---------------------------------------------------------------------------
__device__ __forceinline__ void eliminate_cols(float* __restrict__ A, int cstart,
                                               int cendExcl, float* __restrict__ pivs,
                                               int tid, float* s_inv, float* s_invs,
                                               float (*s_Lm)[16]) {
  int c = cstart;
  while (c < cendExcl) {
    if (((c & 15) == 0) && (cendExcl - c >= 16)) {
      eliminate_panel16(A, c, pivs, tid, s_invs, s_Lm);
      c += 16;
    } else {
      eliminate_one_col(A, c, pivs, tid, s_inv);
      c += 1;
    }
  }
}

// ---------------------------------------------------------------------------
// Kernel 2: sequential sampler, one workgroup (384 threads = 12 wave32).
// Incremental LU with checkpoint reuse (positions strictly increase, each step
// only perturbs diagonal entry (prev,prev)).
// ---------------------------------------------------------------------------
__global__ __launch_bounds__(D) void sampler_kernel(const float* __restrict__ u,
                                                    float* __restrict__ Abase,
                                                    float* __restrict__ Awork,
                                                    float* __restrict__ pivs,
                                                    float* __restrict__ probs,
                                                    float* __restrict__ out) {
  __shared__ float s_inv;
  __shared__ float s_invs[16];
  __shared__ float s_Lm[16][16];
  __shared__ int s_pos;
  const int tid = threadIdx.x;
  int prev = -1;

  for (int k = 0; k < NN; ++k) {
    const int cbase = (prev < 0) ? 0 : prev;
    if (k > 0 && tid == 0) Abase[prev * D + prev] -= 1.0f;  // occ update
    __syncthreads();

    const int xmax = D - NN + k + 1;
    const int cend = xmax - 1;  // eliminate columns [cbase, cend)

    // Coalesced full-row copy of the live part: Abase -> Awork.
    for (int r = cbase; r < D; ++r) Awork[r * D + tid] = Abase[r * D + tid];
    __syncthreads();

    eliminate_cols(Awork, cbase, cend, pivs, tid, &s_inv, s_invs, s_Lm);

    if (tid == 0) {
      pivs[cend] = Awork[cend * D + cend];

      const int xmin = prev + 1;
      float cp = 1.0f, total = 0.0f;
      for (int i = xmin; i < xmax; ++i) {
        float p = cp * (1.0f - pivs[i]);
        if (fabsf(p) <= 1e-15f) p = 0.0f;
        probs[i] = p;
        total += p;
        cp *= pivs[i];
      }
      const float thr = u[k] * total;
      int cnt = (thr > 0.0f) ? xmin : 0;
      float run = 0.0f;
      for (int i = xmin; i < xmax; ++i) {
        run += probs[i];
        if (run < thr) ++cnt;
      }
      const int pos = (cnt < xmax - 1) ? cnt : (xmax - 1);
      const float cond_p = (pos >= xmin && pos < xmax) ? probs[pos] : 0.0f;
      out[k] = (float)pos;
      out[NN + k] = cond_p;
      s_pos = pos;
    }
    __syncthreads();
    const int pos = s_pos;

    // Advance checkpoint to column pos.
    eliminate_cols(Abase, cbase, pos, nullptr, tid, &s_inv, s_invs, s_Lm);
    prev = pos;
  }
}

// ---------------------------------------------------------------------------
extern "C" void kernel_launch(void* const* d_in, const int* in_sizes, int n_in,
                              void* d_out, int out_size, void* d_ws, size_t ws_size,
                              hipStream_t stream) {
  (void)in_sizes; (void)n_in; (void)out_size; (void)ws_size;
  const float* eig = (const float*)d_in[0];  // 384x384 row-major
  const float* u   = (const float*)d_in[1];  // 96 uniforms

  float* ws    = (float*)d_ws;
  float* Abase = ws;              // D*D
  float* Awork = ws + D * D;      // D*D
  float* pivs  = ws + 2 * D * D;  // D
  float* probs = pivs + D;        // D
  float* out   = (float*)d_out;   // [positions(96), cond_probs(96)]

  dim3 g1(D / 16, D / 16);
  build_G_kernel<<<g1, 32, 0, stream>>>(eig, Abase);
  sampler_kernel<<<1, D, 0, stream>>>(u, Abase, Awork, pivs, probs, out);
}